// MultiHeadedAttention_24592982737253
// MI455X (gfx1250) — compile-verified
//
#include <hip/hip_runtime.h>
#include <hip/hip_bf16.h>
#include <math.h>

typedef __attribute__((ext_vector_type(2))) float v2f;
typedef __attribute__((ext_vector_type(8))) float v8f;

#define TD_STRIDE 131072   // D*R*F = 256*64*8 : stride over t in the [T,D,R,F] layout
// input/output addr(t, dm, r, f) = t*TD_STRIDE + dm*512 + r*8 + f

// ---------------- Kernel 1: xv_mean[g][dm] = mean over (t,r) of v[t,dm,r,g] ----
__global__ void k_vmean_reduce(const float* __restrict__ v, float* __restrict__ xv_mean) {
    int idx = blockIdx.x * blockDim.x + threadIdx.x;   // 2048 threads
    int g = idx >> 8, dm = idx & 255;
    float s = 0.f;
    const float* p = v + dm * 512 + g;
    for (int t = 0; t < 8; ++t)
        for (int r = 0; r < 64; ++r)
            s += p[t * TD_STRIDE + r * 8];
    xv_mean[g * 256 + dm] = s * (1.f / 512.f);
}

// ---------------- Kernel 2: V_mean[g][j] = xv_mean[g,:] @ Wv[j,:] + bv[j] ------
__global__ void k_vmean_proj(const float* __restrict__ xv_mean, const float* __restrict__ Wv,
                             const float* __restrict__ bv, float* __restrict__ V_mean) {
    int j = threadIdx.x;                                // 256 threads, 1 block
    for (int g = 0; g < 8; ++g) {
        float acc = bv[j];
        const float* xm = xv_mean + g * 256;
        const float* w  = Wv + j * 256;
        for (int dm = 0; dm < 256; ++dm) acc += xm[dm] * w[dm];
        V_mean[g * 256 + j] = acc;
    }
}

// ---------------- Kernel 3: Q/K projection GEMM via V_WMMA_F32_16X16X4_F32 -----
// C[m, j] = sum_dm X[m, dm] * W[j, dm] + b[j], written to out[t, j, r, f]
__global__ void __launch_bounds__(32) k_proj_qk(
        const float* __restrict__ Xq, const float* __restrict__ Xk,
        const float* __restrict__ Wq, const float* __restrict__ bq,
        const float* __restrict__ Wk, const float* __restrict__ bk,
        float* __restrict__ out) {
    int bid   = blockIdx.x;
    int which = bid >> 12;                 // 0 = q, 1 = k
    int tile  = bid & 4095;
    int m0 = (tile >> 4) << 4;             // 256 M-tiles
    int n0 = (tile & 15) << 4;             // 16  N-tiles
    const float* X    = which ? Xk : Xq;
    const float* W    = which ? Wk : Wq;
    const float* bias = which ? bk : bq;
    float* O = out + which * 1048576;

    int lane = threadIdx.x;
    int lm = lane & 15;
    int lk = (lane >> 4) << 1;             // 0 or 2

    int m  = m0 + lm;
    int t  = m >> 9;
    int rf = m & 511;
    const float* xrow = X + t * TD_STRIDE + rf;   // element dm at xrow[dm*512]
    const float* wrow = W + (n0 + lm) * 256;      // element k  at wrow[k]

    v8f c = {};
    for (int kb = 0; kb < 256; kb += 4) {
        v2f a, b;
        a.x = xrow[(kb + lk) * 512];
        a.y = xrow[(kb + lk + 1) * 512];
        b.x = wrow[kb + lk];
        b.y = wrow[kb + lk + 1];
        c = __builtin_amdgcn_wmma_f32_16x16x4_f32(false, a, false, b, (short)0, c, false, false);
    }
    float bj = bias[n0 + lm];
    int half = lane >> 4;
    for (int v = 0; v < 8; ++v) {
        int mm = m0 + v + 8 * half;        // row M of C held by this lane/vgpr
        int tt = mm >> 9;
        int rr = mm & 511;
        O[tt * TD_STRIDE + (n0 + lm) * 512 + rr] = c[v] + bj;
    }
}

// ---------------- Kernel 4: fused scores (WMMA) -> softmax(Tk,Fk) -> pool ------
__global__ void __launch_bounds__(256) k_attn(const float* __restrict__ out,
                                              float* __restrict__ p_mean) {
    __shared__ float sQ[16 * 128];         // staged Q tile
    __shared__ float sS[8 * 16 * 64];      // per-wave 16x64 score tile
    __shared__ float sP[8 * 16 * 8];       // per-wave partial p sums

    int hh = blockIdx.x >> 8;              // head
    int m0 = (blockIdx.x & 255) << 4;      // query tile base
    int tid  = threadIdx.x;
    int wave = tid >> 5;
    int lane = tid & 31;
    int lm = lane & 15;
    int lk = (lane >> 4) << 1;

    const float* Q = out;                  // q_out region, layout [t, j, r, f]
    const float* K = out + 1048576;        // k_out region

    // stage the 16x128 Q tile for this head into LDS
    for (int i = tid; i < 2048; i += 256) {
        int qq = i >> 7, dd = i & 127;
        int m = m0 + qq;
        int t = m >> 9, rf = m & 511;
        sQ[qq * 128 + dd] = Q[t * TD_STRIDE + (hh * 128 + dd) * 512 + rf];
    }
    __syncthreads();

    const float inv_sqrt_d = 0.08838834764831845f;   // 1/sqrt(128)
    float accg[8];
    for (int g = 0; g < 8; ++g) accg[g] = 0.f;

    float* myS = sS + wave * 1024;

    // key-local index j = s*8 + g; for N-tile nt: s = 2*nt + (lm>>3), g = lm&7
    int sg = lm >> 3, gg = lm & 7;

    for (int ui = 0; ui < 8; ++ui) {
        int u = wave * 8 + ui;             // Rk value owned this iteration
        // key row (s,u,g) address: s*TD_STRIDE + (hh*128+dd)*512 + u*8 + g
        const float* kp0 = K + sg * TD_STRIDE + (hh * 128) * 512 + u * 8 + gg;
        const float* kp1 = kp0 + 2 * TD_STRIDE;
        const float* kp2 = kp0 + 4 * TD_STRIDE;
        const float* kp3 = kp0 + 6 * TD_STRIDE;

        v8f c0 = {}, c1 = {}, c2 = {}, c3 = {};
        for (int kb = 0; kb < 128; kb += 4) {
            v2f a, b;
            a.x = sQ[lm * 128 + kb + lk];
            a.y = sQ[lm * 128 + kb + lk + 1];
            int o0 = (kb + lk) * 512, o1 = (kb + lk + 1) * 512;
            b.x = kp0[o0]; b.y = kp0[o1];
            c0 = __builtin_amdgcn_wmma_f32_16x16x4_f32(false, a, false, b, (short)0, c0, false, false);
            b.x = kp1[o0]; b.y = kp1[o1];
            c1 = __builtin_amdgcn_wmma_f32_16x16x4_f32(false, a, false, b, (short)0, c1, false, false);
            b.x = kp2[o0]; b.y = kp2[o1];
            c2 = __builtin_amdgcn_wmma_f32_16x16x4_f32(false, a, false, b, (short)0, c2, false, false);
            b.x = kp3[o0]; b.y = kp3[o1];
            c3 = __builtin_amdgcn_wmma_f32_16x16x4_f32(false, a, false, b, (short)0, c3, false, false);
        }
        // scatter scaled scores to this wave's LDS tile: row M, col j
        int half = lane >> 4;
        for (int v = 0; v < 8; ++v) {
            int mq = v + 8 * half;
            myS[mq * 64 + lm]      = c0[v] * inv_sqrt_d;
            myS[mq * 64 + 16 + lm] = c1[v] * inv_sqrt_d;
            myS[mq * 64 + 32 + lm] = c2[v] * inv_sqrt_d;
            myS[mq * 64 + 48 + lm] = c3[v] * inv_sqrt_d;
        }
        __syncthreads();
        // softmax over the 64 (s,g) entries per query; accumulate per-g sums
        if (lane < 16) {
            float* row = myS + lane * 64;
            float mx = row[0];
            for (int jj = 1; jj < 64; ++jj) mx = fmaxf(mx, row[jj]);
            float eg[8]; for (int g = 0; g < 8; ++g) eg[g] = 0.f;
            float Z = 0.f;
            for (int jj = 0; jj < 64; ++jj) {
                float e = __expf(row[jj] - mx);
                Z += e;
                eg[jj & 7] += e;
            }
            float rZ = 1.f / Z;
            for (int g = 0; g < 8; ++g) accg[g] += eg[g] * rZ;
        }
        __syncthreads();
    }

    if (lane < 16)
        for (int g = 0; g < 8; ++g) sP[wave * 128 + lane * 8 + g] = accg[g];
    __syncthreads();
    if (tid < 128) {
        int qq = tid >> 3, g = tid & 7;
        float s = 0.f;
        for (int w = 0; w < 8; ++w) s += sP[w * 128 + qq * 8 + g];
        p_mean[hh * 32768 + (m0 + qq) * 8 + g] = s * (1.f / 512.f);
    }
}

// ---------------- Kernel 5: x_out[t,j,r,f<4] = sum_g p_mean * V_mean ----------
__global__ void k_out(const float* __restrict__ p_mean, const float* __restrict__ V_mean,
                      float* __restrict__ x_out) {
    int idx = blockIdx.x * blockDim.x + threadIdx.x;   // 524288
    int f = idx & 3;
    int r = (idx >> 2) & 63;
    int j = (idx >> 8) & 255;
    int t = idx >> 16;
    int n = t * 512 + r * 8 + f;
    int hh = j >> 7;
    const float* p = p_mean + hh * 32768 + n * 8;
    float acc = 0.f;
    for (int g = 0; g < 8; ++g) acc += p[g] * V_mean[g * 256 + j];
    x_out[idx] = acc;
}

extern "C" void kernel_launch(void* const* d_in, const int* in_sizes, int n_in,
                              void* d_out, int out_size, void* d_ws, size_t ws_size,
                              hipStream_t stream) {
    const float* q  = (const float*)d_in[0];
    const float* k  = (const float*)d_in[1];
    const float* v  = (const float*)d_in[2];
    const float* Wq = (const float*)d_in[3];
    const float* bq = (const float*)d_in[4];
    const float* Wk = (const float*)d_in[5];
    const float* bk = (const float*)d_in[6];
    const float* Wv = (const float*)d_in[7];
    const float* bv = (const float*)d_in[8];
    float* out = (float*)d_out;          // [q_out | k_out | x_out]
    float* ws  = (float*)d_ws;
    float* xv_mean = ws;                 // 2048 floats
    float* V_mean  = ws + 2048;          // 2048 floats
    float* p_mean  = ws + 4096;          // 65536 floats  (~280 KB total ws)

    k_vmean_reduce<<<8, 256, 0, stream>>>(v, xv_mean);
    k_vmean_proj<<<1, 256, 0, stream>>>(xv_mean, Wv, bv, V_mean);
    k_proj_qk<<<8192, 32, 0, stream>>>(q, k, Wq, bq, Wk, bk, out);
    k_attn<<<512, 256, 0, stream>>>(out, p_mean);
    k_out<<<2048, 256, 0, stream>>>(p_mean, V_mean, out + 2097152);
}